// TriggerGenerator_120259084719
// MI455X (gfx1250) — compile-verified
//
#include <hip/hip_runtime.h>
#include <cstdint>
#include <cstddef>

#define CIN 256
#define HID 64

typedef __attribute__((ext_vector_type(2))) float v2f;
typedef __attribute__((ext_vector_type(8))) float v8f;
typedef __attribute__((ext_vector_type(4))) int v4i;

// ---- CDNA5 async global->LDS staging (ASYNCcnt path), guarded ----
#if defined(__has_builtin)
#  if __has_builtin(__builtin_amdgcn_global_load_async_to_lds_b128) && \
      __has_builtin(__builtin_amdgcn_s_wait_asynccnt)
#    define USE_ASYNC_LDS 1
#  endif
#endif
#ifndef USE_ASYNC_LDS
#  define USE_ASYNC_LDS 0
#endif

#if USE_ASYNC_LDS
typedef __attribute__((address_space(1))) v4i* as1_v4i_ptr;
typedef __attribute__((address_space(3))) v4i* as3_v4i_ptr;
#endif

__device__ __forceinline__ void stage16(const float* __restrict__ g, float* l) {
#if USE_ASYNC_LDS
  __builtin_amdgcn_global_load_async_to_lds_b128(
      (as1_v4i_ptr)g, (as3_v4i_ptr)l, 0, 0);
#else
  *(float4*)l = *(const float4*)g;
#endif
}

__device__ __forceinline__ void stage_fence() {
#if USE_ASYNC_LDS
  __builtin_amdgcn_s_wait_asynccnt(0);
#endif
  __syncthreads();
}

// ===================== WMMA GEMM: C[M,Ncols] = A[M,KTOT] @ B[KTOT,Ncols] =====================
// Block = 256 threads (8 waves). Block tile = 128 rows x 64 cols. K staged in 64-chunks in LDS.
// Each wave computes a 16x64 strip with 4 independent f32 16x16x4 WMMA accumulator chains.
// EPI==1: fused  out = tanh(acc + bias[col]).
template <int KTOT, int EPI>
__global__ __launch_bounds__(256) void gemm_wmma_kernel(
    const float* __restrict__ A, const float* __restrict__ B,
    const float* __restrict__ bias, float* __restrict__ C, int M, int Ncols) {
  constexpr int KC = 64;
  constexpr int ASTR = KC + 4;  // pad so A-frag reads are LDS-bank-conflict free, 16B aligned
  __shared__ __align__(16) float As[128 * ASTR];
  __shared__ __align__(16) float Bs[KC * 64];

  const int tid = threadIdx.x;
  const int lane = tid & 31;
  const int wave = tid >> 5;
  const int l15 = lane & 15;
  const int hi = lane >> 4;  // 0 or 1 (half-wave)
  const int row0 = blockIdx.x * 128;
  const int col0 = blockIdx.y * 64;

  v8f acc[4] = {v8f{}, v8f{}, v8f{}, v8f{}};

  for (int kc = 0; kc < KTOT; kc += KC) {
    // Stage A chunk: 128x64 floats (rows beyond M are clamped; they only feed
    // output rows beyond M, which are never stored).
#pragma unroll
    for (int j = 0; j < 8; ++j) {
      int idx = j * 256 + tid;          // float4 slot 0..2047
      int r = idx >> 4;                 // 0..127
      int c4 = (idx & 15) << 2;         // 0..60
      int grow = row0 + r;
      grow = grow < M ? grow : (M - 1);
      stage16(&A[(size_t)grow * KTOT + kc + c4], &As[r * ASTR + c4]);
    }
    // Stage B chunk: 64x64 floats
#pragma unroll
    for (int j = 0; j < 4; ++j) {
      int idx = j * 256 + tid;          // float4 slot 0..1023
      int r = idx >> 4;                 // 0..63
      int c4 = (idx & 15) << 2;
      stage16(&B[(size_t)(kc + r) * Ncols + col0 + c4], &Bs[r * 64 + c4]);
    }
    stage_fence();

    const float* aRow = &As[(wave * 16 + l15) * ASTR];
#pragma unroll
    for (int k = 0; k < KC; k += 4) {
      // A-frag (16x4 f32): lanes 0-15 -> K=k..k+1, lanes 16-31 -> K=k+2..k+3
      v2f a = *(const v2f*)&aRow[k + hi * 2];
#pragma unroll
      for (int n = 0; n < 4; ++n) {
        // B-frag (4x16 f32): VGPR0 K=k(+2 for hi half), VGPR1 K=k+1(+2)
        v2f b;
        b.x = Bs[(k + hi * 2 + 0) * 64 + n * 16 + l15];
        b.y = Bs[(k + hi * 2 + 1) * 64 + n * 16 + l15];
        acc[n] = __builtin_amdgcn_wmma_f32_16x16x4_f32(
            false, a, false, b, (short)0, acc[n], false, false);
      }
    }
    __syncthreads();
  }

  // Store (C/D layout: VGPR v -> row v (+8 for hi half), col = lane%16)
#pragma unroll
  for (int n = 0; n < 4; ++n) {
    int c = col0 + n * 16 + l15;
#pragma unroll
    for (int v = 0; v < 8; ++v) {
      int r = row0 + wave * 16 + v + hi * 8;
      if (r < M) {
        float val = acc[n][v];
        if (EPI == 1) val = tanhf(val + bias[c]);
        C[(size_t)r * Ncols + c] = val;
      }
    }
  }
}

// ===================== support kernels =====================
__global__ void k_fill(float* __restrict__ p, float v, int n) {
  int i = blockIdx.x * blockDim.x + threadIdx.x;
  if (i < n) p[i] = v;
}

__global__ void k_deg(const long long* __restrict__ dst, float* __restrict__ deg, int E) {
  int e = blockIdx.x * blockDim.x + threadIdx.x;
  if (e < E) atomicAdd(&deg[(int)dst[e]], 1.0f);
}

__global__ void k_rsqrt_inplace(float* __restrict__ d, int n) {
  int i = blockIdx.x * blockDim.x + threadIdx.x;
  if (i < n) d[i] = rsqrtf(d[i]);  // deg >= 1 always (self loop)
}

// Scatter-add message passing over 64 features: 16 lanes (float4 each) per edge.
// h rows live in L2 (25.6 MB), atomics resolve in L2.
__global__ void k_aggregate64(const float* __restrict__ h,
                              const long long* __restrict__ src,
                              const long long* __restrict__ dst,
                              const float* __restrict__ dinv,
                              float* __restrict__ out, int E) {
  int t = blockIdx.x * blockDim.x + threadIdx.x;
  int e = t >> 4;
  if (e >= E) return;
  int f4 = (t & 15) << 2;
  int s = (int)src[e];
  int d = (int)dst[e];
  float norm = dinv[s] * dinv[d];
  float4 v = *(const float4*)&h[(size_t)s * HID + f4];
  float* o = &out[(size_t)d * HID + f4];
  atomicAdd(o + 0, v.x * norm);
  atomicAdd(o + 1, v.y * norm);
  atomicAdd(o + 2, v.z * norm);
  atomicAdd(o + 3, v.w * norm);
}

// layer-1 epilogue: a = relu(a + h*dinv^2 + b1[f])   (self-loop folded analytically)
__global__ void k_self_bias_relu(float* __restrict__ a, const float* __restrict__ h,
                                 const float* __restrict__ dinv,
                                 const float* __restrict__ b, int N) {
  int t = blockIdx.x * blockDim.x + threadIdx.x;
  if (t >= N * HID) return;
  int i = t >> 6;
  int f = t & (HID - 1);
  float di = dinv[i];
  float val = a[t] + h[t] * di * di + b[f];
  a[t] = val > 0.f ? val : 0.f;
}

// layer-2 pre-GEMM epilogue: g += a*dinv^2   (self loop; Agg(a)@W2 == Agg(a@W2))
__global__ void k_self_add(float* __restrict__ g, const float* __restrict__ a,
                           const float* __restrict__ dinv, int N) {
  int t = blockIdx.x * blockDim.x + threadIdx.x;
  if (t >= N * HID) return;
  int i = t >> 6;
  float di = dinv[i];
  g[t] += a[t] * di * di;
}

// ===================== launch =====================
extern "C" void kernel_launch(void* const* d_in, const int* in_sizes, int n_in,
                              void* d_out, int out_size, void* d_ws, size_t ws_size,
                              hipStream_t stream) {
  const float* x = (const float*)d_in[0];
  const long long* ei = (const long long*)d_in[1];
  const float* W1 = (const float*)d_in[2];
  const float* b1 = (const float*)d_in[3];
  const float* W2 = (const float*)d_in[4];
  const float* b2 = (const float*)d_in[5];
  float* out = (float*)d_out;
  (void)n_in; (void)out_size; (void)ws_size;

  const int N = in_sizes[0] / CIN;   // 100000 (multiple of 16)
  const int E = in_sizes[1] / 2;     // 1600000
  const long long* srcs = ei;
  const long long* dsts = ei + E;

  char* ws = (char*)d_ws;
  size_t off = 0;
  auto walloc = [&](size_t bytes) -> void* {
    void* p = ws + off;
    off += (bytes + 255) & ~(size_t)255;
    return p;
  };
  float* dinv = (float*)walloc((size_t)N * sizeof(float));
  float* h1 = (float*)walloc((size_t)N * HID * sizeof(float));   // x @ W1
  float* a1 = (float*)walloc((size_t)N * HID * sizeof(float));   // relu(Agg(h1)+b1)
  float* g2 = (float*)walloc((size_t)N * HID * sizeof(float));   // Agg(a1)

  const int T = 256;
  const int NF = N * HID;

  // deg -> dinv (self loop contributes the init 1.0)
  k_fill<<<(N + T - 1) / T, T, 0, stream>>>(dinv, 1.0f, N);
  k_deg<<<(E + T - 1) / T, T, 0, stream>>>(dsts, dinv, E);
  k_rsqrt_inplace<<<(N + T - 1) / T, T, 0, stream>>>(dinv, N);

  // h1 = x @ W1
  dim3 grid1((N + 127) / 128, HID / 64);
  gemm_wmma_kernel<CIN, 0><<<grid1, 256, 0, stream>>>(x, W1, nullptr, h1, N, HID);

  // a1 = relu(Agg(h1) + self + b1)
  k_fill<<<(NF + T - 1) / T, T, 0, stream>>>(a1, 0.0f, NF);
  k_aggregate64<<<((E * 16) + T - 1) / T, T, 0, stream>>>(h1, srcs, dsts, dinv, a1, E);
  k_self_bias_relu<<<(NF + T - 1) / T, T, 0, stream>>>(a1, h1, dinv, b1, N);

  // g2 = Agg(a1) + self   (aggregate in 64-dim hidden space: 4x less traffic)
  k_fill<<<(NF + T - 1) / T, T, 0, stream>>>(g2, 0.0f, NF);
  k_aggregate64<<<((E * 16) + T - 1) / T, T, 0, stream>>>(a1, srcs, dsts, dinv, g2, E);
  k_self_add<<<(NF + T - 1) / T, T, 0, stream>>>(g2, a1, dinv, N);

  // out = tanh(g2 @ W2 + b2)  (bias + tanh fused into GEMM epilogue)
  dim3 grid2((N + 127) / 128, CIN / 64);
  gemm_wmma_kernel<HID, 1><<<grid2, 256, 0, stream>>>(g2, W2, b2, out, N, CIN);
}